// GroupTokenBuilder_73847667687685
// MI455X (gfx1250) — compile-verified
//
#include <hip/hip_runtime.h>
#include <hip/hip_bf16.h>
#include <math.h>

typedef __attribute__((ext_vector_type(2))) float v2f;
typedef __attribute__((ext_vector_type(8))) float v8f;

constexpr int   B_   = 32;
constexpr int   A_   = 1024;
constexpr int   D_   = 128;
constexpr float THR2 = 0.05f * 0.05f;

// ---------------------------------------------------------------------------
// 1) Greedy sequential proximity clustering. One 1024-thread workgroup per
//    batch; the i-scan is sequential, the per-j "take" is parallel. Barriers
//    only on iterations that start a new group (uniform branch), so the
//    barrier count scales with #groups (~hundreds), not A (1024).
// ---------------------------------------------------------------------------
__global__ __launch_bounds__(1024) void cluster_kernel(
    const float* __restrict__ coords,      // [B, A, 2]
    const int*   __restrict__ amask,       // [B, A]
    int*         __restrict__ assigned_ws, // [B, A]
    float*       __restrict__ a2g_out)     // [B, A] (float-encoded int)
{
    const int b = blockIdx.x;
    const int j = threadIdx.x;

    __shared__ float cx[A_];
    __shared__ float cy[A_];
    __shared__ int   asg[A_];

    const float xj = coords[(b * A_ + j) * 2 + 0];
    const float yj = coords[(b * A_ + j) * 2 + 1];
    int myg = amask[b * A_ + j] ? -1 : -2;   // -1 = valid & unassigned, -2 = invalid

    cx[j]  = xj;
    cy[j]  = yj;
    asg[j] = myg;
    __syncthreads();

    int gid = 0;
    for (int i = 0; i < A_; ++i) {
        const int ai = asg[i];               // uniform read (no writes in flight)
        if (ai == -1) {                      // block-uniform branch
            const float dx = xj - cx[i];
            const float dy = yj - cy[i];
            const bool take = (myg == -1) && (dx * dx + dy * dy < THR2);
            __syncthreads();                 // all reads done before any write
            if (take) { myg = gid; asg[j] = gid; }
            ++gid;
            __syncthreads();                 // writes visible to next iteration
        }
    }

    assigned_ws[b * A_ + j] = myg;
    a2g_out[b * A_ + j]     = (float)(myg >= 0 ? myg : -1);
}

// ---------------------------------------------------------------------------
// 2) Deterministic segment-mean pooling: one 128-thread block per (b, group)
//    output row; ordered gather over the 1024 agents (no float atomics, so
//    results are bitwise stable across graph replays). The per-batch
//    `assigned` row (4 KB) is staged into LDS once, so the inner-loop lookup
//    is a same-address LDS broadcast instead of 1024 redundant VMEM loads.
// ---------------------------------------------------------------------------
__global__ __launch_bounds__(128) void pool_kernel(
    const float* __restrict__ embeds,      // [B, A, D]
    const int*   __restrict__ assigned_ws, // [B, A]
    float*       __restrict__ tokens,      // [B, A, D] (means out)
    float*       __restrict__ mask_out)    // [B, A]
{
    const int row = blockIdx.x;            // b*A + g
    const int b   = row >> 10;
    const int g   = row & (A_ - 1);
    const int d   = threadIdx.x;

    __shared__ int asg_s[A_];
    {
        const int* asg_g = assigned_ws + b * A_;
        #pragma unroll
        for (int t = 0; t < A_ / 128; ++t)
            asg_s[t * 128 + d] = asg_g[t * 128 + d];
    }
    __syncthreads();

    const float* eb = embeds + (size_t)b * A_ * D_;

    float acc = 0.0f;
    int   cnt = 0;
    for (int jj = 0; jj < A_; ++jj) {
        if (asg_s[jj] == g) {              // uniform predicate, LDS broadcast
            acc += eb[(size_t)jj * D_ + d];
            ++cnt;
        }
    }

    const float inv = 1.0f / fmaxf((float)cnt, 1.0f);
    tokens[(size_t)row * D_ + d] = acc * inv;
    if (d == 0) mask_out[row] = (cnt > 0) ? 1.0f : 0.0f;
}

// ---------------------------------------------------------------------------
// 3) Fused MLP: out = GELU(X @ W1 + b1) @ W2 + b2, in-place on the token
//    tensor (each 16-row tile only depends on itself). 8 waves per block,
//    wave w computes output columns [16w, 16w+16) via V_WMMA_F32_16X16X4_F32
//    (exact fp32).
//    LDS tiles padded to stride 132 floats -> conflict-free column reads
//    across the 64 LDS banks; compiler fuses A-fragment reads into
//    ds_load_2addr_b64 and clauses the W column loads.
// ---------------------------------------------------------------------------
__global__ __launch_bounds__(256) void mlp_kernel(
    float*       __restrict__ tokens,  // [B*A, D], read+write in place
    const float* __restrict__ W1,      // [D, D] row-major (k, n)
    const float* __restrict__ b1,      // [D]
    const float* __restrict__ W2,      // [D, D]
    const float* __restrict__ b2)      // [D]
{
    __shared__ float Xs[16][132];
    __shared__ float Hs[16][132];

    const int row0 = blockIdx.x * 16;
    const int tid  = threadIdx.x;

    // Cooperative coalesced load of the 16x128 input tile.
    for (int t = tid; t < 16 * 128; t += 256) {
        const int r = t >> 7;
        const int c = t & 127;
        Xs[r][c] = tokens[(size_t)(row0 + r) * D_ + c];
    }
    __syncthreads();

    const int wv     = tid >> 5;           // wave 0..7 -> output column block
    const int lane   = tid & 31;
    const int m      = lane & 15;          // A-matrix row / C column-lane
    const int kh     = (lane >> 4) << 1;   // K sub-offset: lanes 16-31 hold K+2
    const int rowoff = (lane >> 4) << 3;   // C rows: lanes 16-31 hold M+8
    const int n      = (wv << 4) + m;      // output column 0..127

    // ---- GEMM 1: H = GELU(X @ W1 + b1) ----
    v8f acc = {};
    #pragma unroll
    for (int k = 0; k < D_; k += 4) {
        v2f a, bf;
        a[0]  = Xs[m][k + kh];
        a[1]  = Xs[m][k + kh + 1];
        bf[0] = W1[(size_t)(k + kh)     * D_ + n];
        bf[1] = W1[(size_t)(k + kh + 1) * D_ + n];
        acc = __builtin_amdgcn_wmma_f32_16x16x4_f32(
                  false, a, false, bf, (short)0, acc, false, false);
    }
    {
        const float bias = b1[n];
        #pragma unroll
        for (int r = 0; r < 8; ++r) {
            const float x = acc[r] + bias;
            // exact GELU: 0.5*x*(1+erf(x/sqrt(2)))
            Hs[r + rowoff][n] = 0.5f * x * (1.0f + erff(x * 0.70710678118654752f));
        }
    }
    __syncthreads();

    // ---- GEMM 2: OUT = H @ W2 + b2 ----
    v8f acc2 = {};
    #pragma unroll
    for (int k = 0; k < D_; k += 4) {
        v2f a, bf;
        a[0]  = Hs[m][k + kh];
        a[1]  = Hs[m][k + kh + 1];
        bf[0] = W2[(size_t)(k + kh)     * D_ + n];
        bf[1] = W2[(size_t)(k + kh + 1) * D_ + n];
        acc2 = __builtin_amdgcn_wmma_f32_16x16x4_f32(
                   false, a, false, bf, (short)0, acc2, false, false);
    }
    {
        const float bias = b2[n];
        #pragma unroll
        for (int r = 0; r < 8; ++r) {
            tokens[(size_t)(row0 + r + rowoff) * D_ + n] = acc2[r] + bias;
        }
    }
}

// ---------------------------------------------------------------------------
// Driver
// ---------------------------------------------------------------------------
extern "C" void kernel_launch(void* const* d_in, const int* in_sizes, int n_in,
                              void* d_out, int out_size, void* d_ws, size_t ws_size,
                              hipStream_t stream)
{
    (void)in_sizes; (void)n_in; (void)out_size; (void)ws_size;

    const float* embeds = (const float*)d_in[0];   // [B, A, D]
    const float* coords = (const float*)d_in[1];   // [B, A, 2]
    const int*   amask  = (const int*)  d_in[2];   // [B, A]
    const float* W1     = (const float*)d_in[3];   // [D, D]
    const float* b1     = (const float*)d_in[4];   // [D]
    const float* W2     = (const float*)d_in[5];   // [D, D]
    const float* b2     = (const float*)d_in[6];   // [D]

    float* tokens   = (float*)d_out;                         // B*A*D
    float* mask_out = tokens + (size_t)B_ * A_ * D_;         // B*A
    float* a2g_out  = mask_out + (size_t)B_ * A_;            // B*A

    int* assigned_ws = (int*)d_ws;                           // B*A ints (128 KB)

    cluster_kernel<<<B_, A_, 0, stream>>>(coords, amask, assigned_ws, a2g_out);
    pool_kernel<<<B_ * A_, D_, 0, stream>>>(embeds, assigned_ws, tokens, mask_out);
    mlp_kernel<<<(B_ * A_) / 16, 256, 0, stream>>>(tokens, W1, b1, W2, b2);
}